// EMSA_31619549233461
// MI455X (gfx1250) — compile-verified
//
#include <hip/hip_runtime.h>
#include <hip/hip_bf16.h>

// EMSA fused pipeline for MI455X (gfx1250, wave32).
// Shapes: B=32, C=320, G=5 -> BG=160 groups, CG=64 channels/group, H=W=64, N=4096.

#define BGROUPS 160
#define CG 64
#define HH 64
#define WW 64
#define NPIX 4096

typedef float v2f __attribute__((ext_vector_type(2)));
typedef float v8f __attribute__((ext_vector_type(8)));

// D = A(16x4) * B(4x16) + C, fp32 WMMA (CDNA5).
// Fragment layout (ISA 7.12.2): A: lane%16 = M row, k = vgpr + 2*(lane/16).
// B mirrored: lane%16 = N col, k = vgpr + 2*(lane/16). C/D: n = lane%16, m = vgpr + 8*(lane/16).
__device__ __forceinline__ v8f wmma_f32_4(v2f a, v2f b, v8f c) {
  return __builtin_amdgcn_wmma_f32_16x16x4_f32(false, a, false, b, (short)0, c, false, false);
}

__device__ __forceinline__ float sigmoidf_(float v) {
  return 1.0f / (1.0f + __expf(-v));
}

// ---- CDNA5 async-to-LDS helpers -------------------------------------------
// Flat LDS aperture: addr[31:0] of a generic pointer into LDS is the LDS byte
// offset, which is what the async-load VDST operand wants.
__device__ __forceinline__ uint32_t lds_off_u32(const void* p) {
  return (uint32_t)(uintptr_t)p;
}
// GLOBAL_LOAD_ASYNC_TO_LDS_B128, GVS mode: mem = SADDR64 + VADDR32 (+0).
// ASYNCcnt-tracked DMA straight into LDS (no VGPR round trip).
__device__ __forceinline__ void async_b128(uint32_t ldsoff, const float* sbase,
                                           int voff_bytes) {
  asm volatile("global_load_async_to_lds_b128 %0, %1, %2"
               :: "v"(ldsoff), "v"(voff_bytes), "s"(sbase)
               : "memory");
}
#define WAIT_ASYNCCNT(N) asm volatile("s_wait_asynccnt " #N ::: "memory")

// ---------------------------------------------------------------------------
// Kernel A: coordinate pooling + conv1x1 (WMMA f32) + sigmoid gates.
// One block per bg group. 256 threads.
// ---------------------------------------------------------------------------
__global__ void __launch_bounds__(256) emsa_pool_gate(
    const float* __restrict__ x, const float* __restrict__ w1,
    const float* __restrict__ b1,
    float* __restrict__ gh, float* __restrict__ gw) {
  const int bg = blockIdx.x;
  const int t  = threadIdx.x;

  // LDS: cat[64][130 padded] then scratch region reused as plane[64][65] / w1[64][65]
  __shared__ float smem[64 * 130 + 64 * 65];
  float* cat_l   = smem;             // 8320 floats
  float* scratch = smem + 64 * 130;  // 4160 floats

  const float* gx = x + (size_t)bg * CG * NPIX;

  // ---- pooling: x_h = mean over W, x_w = mean over H ----
  for (int c = 0; c < CG; ++c) {
    __syncthreads();
    const float* p = gx + (size_t)c * NPIX;
    for (int k = 0; k < 16; ++k) {
      int idx = t + 256 * k;
      scratch[(idx >> 6) * 65 + (idx & 63)] = p[idx];  // stride 65: conflict-free
    }
    __syncthreads();
    if (t < 64) {  // row sums -> x_h[c][t]
      float s = 0.f;
      for (int j = 0; j < 64; ++j) s += scratch[t * 65 + j];
      cat_l[c * 130 + t] = s * (1.0f / 64.0f);
    } else if (t < 128) {  // col sums -> x_w[c][j]
      int j = t - 64;
      float s = 0.f;
      for (int i = 0; i < 64; ++i) s += scratch[i * 65 + j];
      cat_l[c * 130 + 64 + j] = s * (1.0f / 64.0f);
    }
  }
  __syncthreads();

  // ---- stage w1 into scratch (stride 65) ----
  for (int i = t; i < 64 * 64; i += 256)
    scratch[(i >> 6) * 65 + (i & 63)] = w1[i];
  __syncthreads();

  // ---- GEMM hw[64,128] = w1[64,64] * cat[64,128], WMMA f32 16x16x4 ----
  const int wv = t >> 5, lane = t & 31;
  const int L = lane & 15, Hh = lane >> 4;
  for (int T = wv; T < 32; T += 8) {  // 4x8 grid of 16x16 tiles, 4 per wave
    const int m0 = (T & 3) << 4;
    const int n0 = (T >> 2) << 4;
    v8f acc = {};
    for (int kk = 0; kk < 64; kk += 4) {
      const int ka = kk + 2 * Hh;
      v2f a, b;
      a.x = scratch[(m0 + L) * 65 + ka];
      a.y = scratch[(m0 + L) * 65 + ka + 1];
      b.x = cat_l[ka * 130 + n0 + L];
      b.y = cat_l[(ka + 1) * 130 + n0 + L];
      acc = wmma_f32_4(a, b, acc);
    }
    for (int v = 0; v < 8; ++v) {
      const int m = m0 + v + 8 * Hh;   // output channel
      const int l = n0 + L;            // position in concat [h | w]
      float r = sigmoidf_(acc[v] + b1[m]);
      if (l < HH)
        gh[((size_t)bg * CG + m) * HH + l] = r;
      else
        gw[((size_t)bg * CG + m) * WW + (l - HH)] = r;
    }
  }
}

// ---------------------------------------------------------------------------
// Kernel B: y = gx * gh * gw; per-channel GroupNorm -> x1; x1 spatial mean.
// One block per (bg, c). 256 threads, deterministic LDS tree reductions.
// ---------------------------------------------------------------------------
__global__ void __launch_bounds__(256) emsa_gate_gn(
    const float* __restrict__ x, const float* __restrict__ gh,
    const float* __restrict__ gw, const float* __restrict__ gamma,
    const float* __restrict__ beta,
    float* __restrict__ x1, float* __restrict__ x1mean) {
  const int id = blockIdx.x;        // bg*64 + c
  const int bg = id >> 6, c = id & 63;
  const int t  = threadIdx.x;

  __shared__ float yb[NPIX];
  __shared__ float ghl[64], gwl[64];
  __shared__ float rs[256], rss[256];

  const float* gxc = x + (size_t)id * NPIX;
  if (t < 64)       ghl[t]      = gh[((size_t)bg * CG + c) * HH + t];
  else if (t < 128) gwl[t - 64] = gw[((size_t)bg * CG + c) * WW + (t - 64)];
  __syncthreads();

  float s = 0.f, ss = 0.f;
  for (int k = 0; k < 16; ++k) {
    int idx = t + 256 * k;
    float yv = gxc[idx] * ghl[idx >> 6] * gwl[idx & 63];
    yb[idx] = yv;
    s += yv; ss += yv * yv;
  }
  rs[t] = s; rss[t] = ss;
  __syncthreads();
  for (int off = 128; off > 0; off >>= 1) {
    if (t < off) { rs[t] += rs[t + off]; rss[t] += rss[t + off]; }
    __syncthreads();
  }
  const float mu   = rs[0] * (1.0f / NPIX);
  const float var  = rss[0] * (1.0f / NPIX) - mu * mu;
  const float rstd = rsqrtf(var + 1e-5f);
  const float g = gamma[c], be = beta[c];

  float* x1c = x1 + (size_t)id * NPIX;
  float s1 = 0.f;
  for (int k = 0; k < 16; ++k) {
    int idx = t + 256 * k;
    float v = (yb[idx] - mu) * rstd * g + be;
    x1c[idx] = v;
    s1 += v;
  }
  __syncthreads();
  rs[t] = s1;
  __syncthreads();
  for (int off = 128; off > 0; off >>= 1) {
    if (t < off) rs[t] += rs[t + off];
    __syncthreads();
  }
  if (t == 0) x1mean[id] = rs[0] * (1.0f / NPIX);
}

// ---------------------------------------------------------------------------
// Kernel W: one-off transpose w3 [64o][576k] -> w3t [576k][64o] so conv tiles
// are contiguous, 16B-aligned async-DMA sources.
// ---------------------------------------------------------------------------
__global__ void __launch_bounds__(256) emsa_w3_transpose(
    const float* __restrict__ w3, float* __restrict__ w3t) {
  int i = blockIdx.x * 256 + threadIdx.x;   // over 576*64
  if (i < 576 * 64) {
    int k = i >> 6, o = i & 63;
    w3t[i] = w3[(size_t)o * 576 + k];
  }
}

// --- conv3x3 staging helpers ------------------------------------------------
// Input rows buffer: [cl*3+r][x+4], stride 72 (16B aligned rows), interior at
// idx 4..67, zero halo at idx 3 (x=-1) and idx 68 (x=64).
__device__ __forceinline__ void stage_in_async(const float* gx, int c0, int y,
                                               float* buf, int t) {
  const uint32_t base = lds_off_u32(buf);
  for (int j = 0; j < 3; ++j) {            // 3 b128 instructions per wave
    int b   = t + 256 * j;                 // 768 b128 tiles total
    int row = b >> 4, seg = b & 15;
    int cl  = row / 3, r = row - cl * 3;
    int yy  = y + r - 1;
    int yc  = yy < 0 ? 0 : (yy > 63 ? 63 : yy);  // clamp: uniform issue count
    int voff = (((c0 + cl) << 12) + (yc << 6) + (seg << 2)) << 2;
    async_b128(base + (uint32_t)(row * 72 + 4 + seg * 4) * 4u, gx, voff);
  }
}
// Overwrite out-of-range rows (clamped loads above) with zeros after the wait.
__device__ __forceinline__ void zerofix_in(int y, float* buf, int t) {
  if (y == 0 || y == 63) {
    int r  = (y == 0) ? 0 : 2;             // 16 rows * 64 floats = 256 float4
    int cl = t >> 4, seg = t & 15;
    float4 z = make_float4(0.f, 0.f, 0.f, 0.f);
    *(float4*)&buf[(cl * 3 + r) * 72 + 4 + seg * 4] = z;
  }
}

// ---------------------------------------------------------------------------
// Kernel C: conv3x3 (SAME, zero pad) as implicit GEMM with WMMA f32.
// One block per (bg, output row y). 8 waves tile M=64 (out ch) x N=64 (pixels);
// K = 576 in 4 chunks of 16 input channels (K_chunk = 144), staged into LDS
// via GLOBAL_LOAD_ASYNC_TO_LDS_B128 with a double-buffered input pipeline.
// ---------------------------------------------------------------------------
__global__ void __launch_bounds__(256) emsa_conv3(
    const float* __restrict__ x, const float* __restrict__ w3t,
    const float* __restrict__ b3, float* __restrict__ x2) {
  const int bg = blockIdx.x >> 6;
  const int y  = blockIdx.x & 63;
  const int t  = threadIdx.x;

  __shared__ __align__(16) float w3l[144 * 64];    // [k_local][o] 36,864 B
  __shared__ __align__(16) float inl[2][48 * 72];  // 2 x 13,824 B

  const float* gx = x + (size_t)bg * CG * NPIX;
  const int wv = t >> 5, lane = t & 31;
  const int L = lane & 15, Hh = lane >> 4;
  const int m0     = (wv & 3) << 4;    // 4 waves across M
  const int n0base = (wv >> 2) << 5;   // 2 waves across N, 2 tiles each

  // halo columns are always zero padding; write once for both buffers
  if (t < 192) {
    int buf = t >= 96, rr = (t & 95) >> 1, side = t & 1;
    inl[buf][rr * 72 + (side ? 68 : 3)] = 0.f;
  }

  // prologue: kick off input chunk 0 DMA
  stage_in_async(gx, 0, y, &inl[0][0], t);

  v8f acc0 = {}, acc1 = {};
  const uint32_t w3base = lds_off_u32(&w3l[0]);

  for (int ci = 0; ci < 4; ++ci) {
    const int c0 = ci * 16;
    __syncthreads();  // previous compute done: safe to overwrite w3l / other buf

    // stage w3t chunk ci: contiguous 9216 floats -> 9 b128 per thread
    {
      const float* src = w3t + (size_t)c0 * 9 * 64;
      for (int j = 0; j < 9; ++j) {
        int i = t + 256 * j;
        async_b128(w3base + (uint32_t)i * 16u, src, i * 16);
      }
    }
    // overlap: kick off next input chunk before waiting
    if (ci < 3) {
      stage_in_async(gx, c0 + 16, y, &inl[(ci + 1) & 1][0], t);
      WAIT_ASYNCCNT(0x3);  // in-order: everything older than the 3 new is done
    } else {
      WAIT_ASYNCCNT(0x0);
    }
    zerofix_in(y, &inl[ci & 1][0], t);
    __syncthreads();

    const float* inb = &inl[ci & 1][0];
    for (int kk = 0; kk < 144; kk += 4) {
      const int ka = kk + 2 * Hh;        // k for vgpr0 of this lane half
      v2f a, b0, b1v;
      a.x = w3l[ka * 64 + m0 + L];
      a.y = w3l[(ka + 1) * 64 + m0 + L];
      // decompose k -> (cl, dy, dx)
      int ca = ka / 9, raa = ka - ca * 9;  int ry0 = raa / 3, rx0 = raa - ry0 * 3;
      int kb = ka + 1;
      int cb = kb / 9, rbb = kb - cb * 9;  int ry1 = rbb / 3, rx1 = rbb - ry1 * 3;
      const int xA = n0base + L + 3;       // array idx of (x + rx - 1) is xA+rx
      b0.x  = inb[(ca * 3 + ry0) * 72 + xA + rx0];
      b0.y  = inb[(cb * 3 + ry1) * 72 + xA + rx1];
      b1v.x = inb[(ca * 3 + ry0) * 72 + xA + 16 + rx0];
      b1v.y = inb[(cb * 3 + ry1) * 72 + xA + 16 + rx1];
      acc0 = wmma_f32_4(a, b0, acc0);
      acc1 = wmma_f32_4(a, b1v, acc1);
    }
  }

  // write x2 = conv + b3
  float* x2row = x2 + (size_t)bg * CG * NPIX + (size_t)y * WW;
  for (int v = 0; v < 8; ++v) {
    const int m  = m0 + v + 8 * Hh;
    const float bv = b3[m];
    x2row[(size_t)m * NPIX + n0base + L]      = acc0[v] + bv;
    x2row[(size_t)m * NPIX + n0base + 16 + L] = acc1[v] + bv;
  }
}

// ---------------------------------------------------------------------------
// Kernel C2: spatial mean of a [.., 4096] plane (deterministic reduction).
// ---------------------------------------------------------------------------
__global__ void __launch_bounds__(256) emsa_mean(
    const float* __restrict__ src, float* __restrict__ dst) {
  const int id = blockIdx.x;
  const int t  = threadIdx.x;
  const float* p = src + (size_t)id * NPIX;
  float s = 0.f;
  for (int k = t; k < NPIX; k += 256) s += p[k];
  __shared__ float rs[256];
  rs[t] = s;
  __syncthreads();
  for (int off = 128; off > 0; off >>= 1) {
    if (t < off) rs[t] += rs[t + off];
    __syncthreads();
  }
  if (t == 0) dst[id] = rs[0] * (1.0f / NPIX);
}

// ---------------------------------------------------------------------------
// Kernel D: channel softmax of pooled means, cross dot products, sigmoid,
// final scaling.  Block = (bg, 1024-pixel slab).
// ---------------------------------------------------------------------------
__global__ void __launch_bounds__(256) emsa_cross(
    const float* __restrict__ x, const float* __restrict__ x1,
    const float* __restrict__ x2, const float* __restrict__ x1mean,
    const float* __restrict__ x2mean, float* __restrict__ out) {
  const int bg   = blockIdx.x >> 2;
  const int pblk = (blockIdx.x & 3) * 1024;
  const int t    = threadIdx.x;

  __shared__ float w11[64], w21[64];
  __shared__ float m1[64], m2[64];
  if (t < 64) { m1[t] = x1mean[bg * 64 + t]; m2[t] = x2mean[bg * 64 + t]; }
  __syncthreads();
  if (t == 0) {  // 64-wide softmaxes, trivially cheap
    float mx1 = -3.4e38f, mx2 = -3.4e38f;
    for (int c = 0; c < 64; ++c) { mx1 = fmaxf(mx1, m1[c]); mx2 = fmaxf(mx2, m2[c]); }
    float s1 = 0.f, s2 = 0.f;
    for (int c = 0; c < 64; ++c) {
      float e1 = __expf(m1[c] - mx1), e2 = __expf(m2[c] - mx2);
      w11[c] = e1; w21[c] = e2; s1 += e1; s2 += e2;
    }
    const float i1 = 1.0f / s1, i2 = 1.0f / s2;
    for (int c = 0; c < 64; ++c) { w11[c] *= i1; w21[c] *= i2; }
  }
  __syncthreads();

  const size_t base = (size_t)bg * CG * NPIX;
  for (int k = 0; k < 4; ++k) {
    const int p = pblk + t + 256 * k;
    float wsum = 0.f;
    for (int c = 0; c < 64; ++c)
      wsum += w11[c] * x2[base + (size_t)c * NPIX + p] +
              w21[c] * x1[base + (size_t)c * NPIX + p];
    const float wt = sigmoidf_(wsum);
    for (int c = 0; c < 64; ++c)
      out[base + (size_t)c * NPIX + p] = x[base + (size_t)c * NPIX + p] * wt;
  }
}

// ---------------------------------------------------------------------------
extern "C" void kernel_launch(void* const* d_in, const int* in_sizes, int n_in,
                              void* d_out, int out_size, void* d_ws, size_t ws_size,
                              hipStream_t stream) {
  const float* x     = (const float*)d_in[0];
  const float* w1    = (const float*)d_in[1];
  const float* b1    = (const float*)d_in[2];
  const float* w3    = (const float*)d_in[3];
  const float* b3    = (const float*)d_in[4];
  const float* gamma = (const float*)d_in[5];
  const float* beta  = (const float*)d_in[6];
  float* out = (float*)d_out;

  // workspace layout (floats)
  float* ws  = (float*)d_ws;
  float* gh  = ws;                                   // 160*64*64
  float* gw  = gh + (size_t)BGROUPS * CG * HH;       // 160*64*64
  float* x1  = gw + (size_t)BGROUPS * CG * WW;       // 160*64*4096
  float* x2  = x1 + (size_t)BGROUPS * CG * NPIX;     // 160*64*4096
  float* x1m = x2 + (size_t)BGROUPS * CG * NPIX;     // 160*64
  float* x2m = x1m + (size_t)BGROUPS * CG;           // 160*64
  float* w3t = x2m + (size_t)BGROUPS * CG;           // 576*64

  emsa_w3_transpose<<<144, 256, 0, stream>>>(w3, w3t);
  emsa_pool_gate<<<BGROUPS, 256, 0, stream>>>(x, w1, b1, gh, gw);
  emsa_gate_gn<<<BGROUPS * CG, 256, 0, stream>>>(x, gh, gw, gamma, beta, x1, x1m);
  emsa_conv3<<<BGROUPS * HH, 256, 0, stream>>>(x, w3t, b3, x2);
  emsa_mean<<<BGROUPS * CG, 256, 0, stream>>>(x2, x2m);
  emsa_cross<<<BGROUPS * 4, 256, 0, stream>>>(x, x1, x2, x1m, x2m, out);
}